// SST_86586540687578
// MI455X (gfx1250) — compile-verified
//
#include <hip/hip_runtime.h>
#include <hip/hip_bf16.h>
#include <math.h>

typedef __attribute__((ext_vector_type(2))) float v2f;
typedef __attribute__((ext_vector_type(8))) float v8f;

// ---------------------------------------------------------------------------
// Generic fp32 WMMA GEMM:  C[M,N] = A[M,K] * B[N,K]^T  (+ bias[n]) (+ D[m,n])
// One wave (32 lanes) per 16x16 output tile. M%16==0, N%16==0, K%4==0.
// Uses V_WMMA_F32_16X16X4_F32 (full fp32 — reference parity, still matrix pipe).
// ---------------------------------------------------------------------------
__global__ __launch_bounds__(32)
void gemm_wmma_f32(const float* __restrict__ A, int lda,
                   const float* __restrict__ B, int ldb,
                   const float* __restrict__ bias,
                   const float* __restrict__ D, int ldd,
                   float* __restrict__ C, int ldc,
                   int K) {
  const int lane  = threadIdx.x & 31;
  const int m0    = blockIdx.y << 4;
  const int n0    = blockIdx.x << 4;
  const int row   = lane & 15;   // M index for A frag / N index for B frag
  const int khalf = lane >> 4;   // which K pair (ISA 16x4 layout)

  const float* Ab = A + (size_t)(m0 + row) * lda + 2 * khalf;
  const float* Bb = B + (size_t)(n0 + row) * ldb + 2 * khalf;

  v8f acc = {0.f, 0.f, 0.f, 0.f, 0.f, 0.f, 0.f, 0.f};

  for (int k = 0; k < K; k += 4) {
    v2f a = *(const v2f*)(Ab + k);   // A[m][k+2*khalf .. +1]
    v2f b = *(const v2f*)(Bb + k);   // B(row n)[k+2*khalf .. +1]
    acc = __builtin_amdgcn_wmma_f32_16x16x4_f32(
        /*neg_a=*/false, a, /*neg_b=*/false, b,
        /*c_mod=*/(short)0, acc, /*reuse_a=*/false, /*reuse_b=*/false);
  }

  // C/D layout: lane 0-15 -> rows 0..7 (vgpr r), lane 16-31 -> rows 8..15
  const int ncol  = n0 + (lane & 15);
  const int rbase = m0 + ((lane >> 4) << 3);
  float badd = bias ? bias[ncol] : 0.f;
#pragma unroll
  for (int r = 0; r < 8; ++r) {
    int mi = rbase + r;
    float v = acc[r] + badd;
    if (D) v += D[(size_t)mi * ldd + ncol];
    C[(size_t)mi * ldc + ncol] = v;
  }
}

__device__ __forceinline__ float sigf(float x) { return 1.f / (1.f + expf(-x)); }

// LSTM cell pointwise update; gates [16,4H], h/c [16,H]; also stores h into
// hs[(m*T + t)*H + j]  (batch-major [N,T,H] like the reference).
__global__ void lstm_cell_kernel(const float* __restrict__ gates,
                                 float* __restrict__ h, float* __restrict__ c,
                                 float* __restrict__ hs, int t, int T, int H) {
  int idx = blockIdx.x * blockDim.x + threadIdx.x;
  if (idx >= 16 * H) return;
  int m = idx / H, j = idx - m * H;
  const float* g = gates + (size_t)m * 4 * H;
  float i  = sigf(g[j]);
  float f  = sigf(g[H + j]);
  float gg = tanhf(g[2 * H + j]);
  float o  = sigf(g[3 * H + j]);
  float cn = f * c[idx] + i * gg;
  c[idx] = cn;
  float hn = o * tanhf(cn);
  h[idx] = hn;
  hs[((size_t)m * T + t) * H + j] = hn;
}

__global__ void add_bias2_kernel(const float* __restrict__ a,
                                 const float* __restrict__ b,
                                 float* __restrict__ o, int n) {
  int i = blockIdx.x * blockDim.x + threadIdx.x;
  if (i < n) o[i] = a[i] + b[i];
}

// scores[bi, c, t] = W_att . tanh(vid_lin[bi,:] + sen_lin[c*Tc+t,:]) + b_att,
// masked to -1e9 for t >= cap_len[c].  One wave per score, A = 256.
__global__ void att_scores_kernel(const float* __restrict__ vid_lin,
                                  const float* __restrict__ sen_lin,
                                  const float* __restrict__ W_att,
                                  const float* __restrict__ b_att,
                                  const int*  __restrict__ cap_len,
                                  float* __restrict__ scores,
                                  int nscores) {
  int wid  = (blockIdx.x * blockDim.x + threadIdx.x) >> 5;
  int lane = threadIdx.x & 31;
  if (wid >= nscores) return;
  int t  = wid % 20;
  int c  = (wid / 20) & 15;
  int bi = wid / (20 * 16);
  const float* vl = vid_lin + (size_t)bi * 256;
  const float* sl = sen_lin + (size_t)(c * 20 + t) * 256;
  float acc = 0.f;
#pragma unroll
  for (int a = lane; a < 256; a += 32)
    acc += W_att[a] * tanhf(vl[a] + sl[a]);
#pragma unroll
  for (int off = 16; off; off >>= 1) acc += __shfl_down(acc, off, 32);
  if (lane == 0) {
    float s = acc + b_att[0];
    int cl = cap_len[c]; if (cl < 1) cl = 1;
    if (t >= cl) s = -1e9f;
    scores[wid] = s;
  }
}

// One wave per (bi, c): softmax over t (lane t holds score), attention-pooled
// caption vector, cosine similarity vs vid_out[bi].
__global__ void att_sim_kernel(const float* __restrict__ scores,
                               const float* __restrict__ cap_out,  // [16,20,512]
                               const float* __restrict__ vid_out,  // [640,512]
                               float* __restrict__ sim,            // [640,16]
                               int npairs) {
  int wid  = (blockIdx.x * blockDim.x + threadIdx.x) >> 5;
  int lane = threadIdx.x & 31;
  if (wid >= npairs) return;
  int c  = wid & 15;
  int bi = wid >> 4;

  const float* sc = scores + (size_t)wid * 20;
  float s = (lane < 20) ? sc[lane] : -3.0e38f;
  float mx = s;
#pragma unroll
  for (int off = 16; off; off >>= 1) mx = fmaxf(mx, __shfl_xor(mx, off, 32));
  float e = (lane < 20) ? expf(s - mx) : 0.f;
  float denom = e;
#pragma unroll
  for (int off = 16; off; off >>= 1) denom += __shfl_xor(denom, off, 32);
  float w = e / denom;

  float w20[20];
#pragma unroll
  for (int t = 0; t < 20; ++t) w20[t] = __shfl(w, t, 32);

  const float* co = cap_out + (size_t)c * 20 * 512;
  const float* vo = vid_out + (size_t)bi * 512;
  float num = 0.f, na2 = 0.f, nb2 = 0.f;
  for (int h = lane; h < 512; h += 32) {
    float a = 0.f;
#pragma unroll
    for (int t = 0; t < 20; ++t) a += w20[t] * co[t * 512 + h];
    float v = vo[h];
    num += a * v; na2 += a * a; nb2 += v * v;
  }
#pragma unroll
  for (int off = 16; off; off >>= 1) {
    num += __shfl_xor(num, off, 32);
    na2 += __shfl_xor(na2, off, 32);
    nb2 += __shfl_xor(nb2, off, 32);
  }
  if (lane == 0)
    sim[wid] = num / fmaxf(sqrtf(na2) * sqrtf(nb2), 1e-6f);
}

// out[b][c] = mean over Tv of sim[(b*40+i)*16 + c]
__global__ void mean_kernel(const float* __restrict__ sim, float* __restrict__ out) {
  int tid = blockIdx.x * blockDim.x + threadIdx.x;
  if (tid >= 256) return;
  int b = tid >> 4, c = tid & 15;
  float s = 0.f;
  for (int i = 0; i < 40; ++i) s += sim[(size_t)(b * 40 + i) * 16 + c];
  out[tid] = s * (1.f / 40.f);
}

extern "C" void kernel_launch(void* const* d_in, const int* in_sizes, int n_in,
                              void* d_out, int out_size, void* d_ws, size_t ws_size,
                              hipStream_t stream) {
  // dims
  const int B = 16, Tv = 40, Bc = 16, Tc = 20;
  const int FC = 2048, VE = 512, H = 512, A = 256, G = 4 * H; // G = 2048

  const float* video_fc   = (const float*)d_in[0];   // [16,40,2048]
  const float* video_cap  = (const float*)d_in[1];   // [16,20,300]
  const int*   cap_length = (const int*)  d_in[2];   // [16]
  const float* W_vemb = (const float*)d_in[3];  const float* b_vemb = (const float*)d_in[4];
  const float* W_wemb = (const float*)d_in[5];  const float* b_wemb = (const float*)d_in[6];
  const float* Wih_v  = (const float*)d_in[7];  const float* Whh_v  = (const float*)d_in[8];
  const float* bih_v  = (const float*)d_in[9];  const float* bhh_v  = (const float*)d_in[10];
  const float* Wih_c  = (const float*)d_in[11]; const float* Whh_c  = (const float*)d_in[12];
  const float* bih_c  = (const float*)d_in[13]; const float* bhh_c  = (const float*)d_in[14];
  const float* W_vid  = (const float*)d_in[15]; const float* b_vid  = (const float*)d_in[16];
  const float* W_sen  = (const float*)d_in[17]; const float* b_sen  = (const float*)d_in[18];
  const float* W_att  = (const float*)d_in[19]; const float* b_att  = (const float*)d_in[20];

  // workspace layout (floats)
  float* ws = (float*)d_ws;
  size_t o = 0;
  auto take = [&](size_t n) { float* p = ws + o; o += n; return p; };
  float* cap_x   = take((size_t)Bc * Tc * VE);   // 320x512
  float* vid_x   = take((size_t)B * Tv * VE);    // 640x512
  float* xW_c    = take((size_t)Bc * Tc * G);    // 320x2048
  float* xW_v    = take((size_t)B * Tv * G);     // 640x2048
  float* gates   = take((size_t)16 * G);         // 16x2048
  float* h_c     = take((size_t)16 * H);
  float* c_c     = take((size_t)16 * H);
  float* h_v     = take((size_t)16 * H);
  float* c_v     = take((size_t)16 * H);
  float* cap_out = take((size_t)Bc * Tc * H);    // 320x512
  float* vid_out = take((size_t)B * Tv * H);     // 640x512
  float* vid_lin = take((size_t)B * Tv * A);     // 640x256
  float* sen_lin = take((size_t)Bc * Tc * A);    // 320x256
  float* scores  = take((size_t)B * Tv * Bc * Tc); // 204800
  float* sim     = take((size_t)B * Tv * Bc);    // 10240
  float* bsum_c  = take((size_t)G);
  float* bsum_v  = take((size_t)G);
  (void)ws_size; (void)in_sizes; (void)n_in; (void)out_size;

  // zero LSTM state (h_c,c_c,h_v,c_v are contiguous)
  hipMemsetAsync(h_c, 0, (size_t)4 * 16 * H * sizeof(float), stream);

  // combined LSTM biases (bih + bhh)
  add_bias2_kernel<<<(G + 255) / 256, 256, 0, stream>>>(bih_c, bhh_c, bsum_c, G);
  add_bias2_kernel<<<(G + 255) / 256, 256, 0, stream>>>(bih_v, bhh_v, bsum_v, G);

  // cap_x = video_caption @ W_wemb^T + b_wemb   [320,300]x[512,300]->[320,512]
  gemm_wmma_f32<<<dim3(VE / 16, (Bc * Tc) / 16), 32, 0, stream>>>(
      video_cap, 300, W_wemb, 300, b_wemb, nullptr, 0, cap_x, VE, 300);
  // vid_x = video_fc @ W_vemb^T + b_vemb       [640,2048]x[512,2048]->[640,512]
  gemm_wmma_f32<<<dim3(VE / 16, (B * Tv) / 16), 32, 0, stream>>>(
      video_fc, FC, W_vemb, FC, b_vemb, nullptr, 0, vid_x, VE, FC);

  // xW precompute: x @ Wih^T + (bih+bhh)  -> [*,2048]
  gemm_wmma_f32<<<dim3(G / 16, (Bc * Tc) / 16), 32, 0, stream>>>(
      cap_x, VE, Wih_c, VE, bsum_c, nullptr, 0, xW_c, G, VE);
  gemm_wmma_f32<<<dim3(G / 16, (B * Tv) / 16), 32, 0, stream>>>(
      vid_x, VE, Wih_v, VE, bsum_v, nullptr, 0, xW_v, G, VE);

  // Caption LSTM: 20 sequential steps (N=16 == one WMMA M-tile)
  for (int t = 0; t < Tc; ++t) {
    gemm_wmma_f32<<<dim3(G / 16, 1), 32, 0, stream>>>(
        h_c, H, Whh_c, H, nullptr, xW_c + (size_t)t * G, Tc * G, gates, G, H);
    lstm_cell_kernel<<<(16 * H + 255) / 256, 256, 0, stream>>>(
        gates, h_c, c_c, cap_out, t, Tc, H);
  }
  // Video LSTM: 40 sequential steps
  for (int t = 0; t < Tv; ++t) {
    gemm_wmma_f32<<<dim3(G / 16, 1), 32, 0, stream>>>(
        h_v, H, Whh_v, H, nullptr, xW_v + (size_t)t * G, Tv * G, gates, G, H);
    lstm_cell_kernel<<<(16 * H + 255) / 256, 256, 0, stream>>>(
        gates, h_v, c_v, vid_out, t, Tv, H);
  }

  // attention projections
  gemm_wmma_f32<<<dim3(A / 16, (B * Tv) / 16), 32, 0, stream>>>(
      vid_out, H, W_vid, H, b_vid, nullptr, 0, vid_lin, A, H);
  gemm_wmma_f32<<<dim3(A / 16, (Bc * Tc) / 16), 32, 0, stream>>>(
      cap_out, H, W_sen, H, b_sen, nullptr, 0, sen_lin, A, H);

  // scores + softmax/att-pool/cosine + mean
  int nscores = B * Tv * Bc * Tc;                 // 204800 (one wave each)
  att_scores_kernel<<<(nscores * 32 + 255) / 256, 256, 0, stream>>>(
      vid_lin, sen_lin, W_att, b_att, cap_length, scores, nscores);
  int npairs = B * Tv * Bc;                       // 10240 (one wave each)
  att_sim_kernel<<<(npairs * 32 + 255) / 256, 256, 0, stream>>>(
      scores, cap_out, vid_out, sim, npairs);
  mean_kernel<<<1, 256, 0, stream>>>(sim, (float*)d_out);
}